// GraphSAGELayer_10892037063139
// MI455X (gfx1250) — compile-verified
//
#include <hip/hip_runtime.h>
#include <hip/hip_bf16.h>
#include <math.h>

typedef __attribute__((ext_vector_type(2))) float v2f;
typedef __attribute__((ext_vector_type(8))) float v8f;

#define IN_DIM  128
#define OUT_DIM 128

// ---------------------------------------------------------------------------
// Phase 1: zero the workspace accumulators (msg_sum [N*128] + deg [N]).
// Must run every call (harness poisons ws once, never re-poisons).
// ---------------------------------------------------------------------------
__global__ void sage_zero_kernel(float* __restrict__ ws, long long n) {
    long long i      = (long long)blockIdx.x * blockDim.x + threadIdx.x;
    long long stride = (long long)gridDim.x * blockDim.x;
    for (; i < n; i += stride) ws[i] = 0.0f;
}

// ---------------------------------------------------------------------------
// Phase 2: edge scatter.  One wave32 per edge: each lane loads float4 of the
// source row (coalesced b128 load) and issues 4 hardware f32 atomics into the
// destination row.  Accumulator + x are L2-resident (51 MB each vs 192 MB L2),
// so the 205M atomics stay on-chip.  Lane 0 bumps the degree counter.
// ---------------------------------------------------------------------------
__global__ void __launch_bounds__(256)
sage_scatter_kernel(const float* __restrict__ x,
                    const int*   __restrict__ ei,   // [2, E] flat, int32
                    float*       __restrict__ msg,  // [N,128] accumulator
                    float*       __restrict__ deg,  // [N]
                    int E) {
    int       lane  = threadIdx.x & 31;
    long long gwave = ((long long)blockIdx.x * blockDim.x + threadIdx.x) >> 5;
    long long nwave = ((long long)gridDim.x  * blockDim.x) >> 5;
    for (long long e = gwave; e < E; e += nwave) {
        int src = ei[e];
        int dst = ei[(long long)E + e];
        float4 v = *((const float4*)(x + (long long)src * IN_DIM) + lane);
        float* p = msg + (long long)dst * IN_DIM + lane * 4;
        unsafeAtomicAdd(p + 0, v.x);
        unsafeAtomicAdd(p + 1, v.y);
        unsafeAtomicAdd(p + 2, v.z);
        unsafeAtomicAdd(p + 3, v.w);
        if (lane == 0) unsafeAtomicAdd(deg + dst, 1.0f);
    }
}

// ---------------------------------------------------------------------------
// Phase 3: fused dual-GEMM + bias + L2-normalize using V_WMMA_F32_16X16X4_F32.
// One wave owns a 16(node)x128(out) tile: 8 v8f accumulators.
//   out = (msg/deg) @ Wl^T + x @ Wr^T + bl ; then row-wise L2 normalize.
// A fragment (16x4 f32): lane l holds A[l&15][k + 2*(l>>4) + {0,1}]
// B fragment (4x16 f32): lane l holds B[k + 2*(l>>4) + {0,1}][l&15],
//   with B[k][n] = W[n][k]  (out = A @ W^T).
// C/D (16x16 f32): VGPR j <-> row j + 8*(l>>4), col l&15.
// ---------------------------------------------------------------------------
__global__ void __launch_bounds__(256)
sage_gemm_kernel(const float* __restrict__ x,
                 const float* __restrict__ msg,
                 const float* __restrict__ deg,
                 const float* __restrict__ Wl,   // [128,128] row-major W[n][k]
                 const float* __restrict__ bl,   // [128]
                 const float* __restrict__ Wr,   // [128,128]
                 float*       __restrict__ out,  // [N,128]
                 int N, int ntiles) {
    int lane = threadIdx.x & 31;
    int wib  = threadIdx.x >> 5;
    int t    = blockIdx.x * 8 + wib;
    if (t >= ntiles) return;           // whole-wave exit; survivors have full EXEC

    const int m0  = t * 16;
    const int hi  = lane >> 4;         // 0: K pair {0,1} / rows 0-7 ; 1: {2,3} / rows 8-15
    const int l15 = lane & 15;

    int rowA = m0 + l15;
    if (rowA >= N) rowA = N - 1;       // safe clamp (N=100000 is 16-aligned anyway)
    const float invd = 1.0f / fmaxf(deg[rowA], 1.0f);

    v8f acc[8];
    v8f zero = {};
#pragma unroll
    for (int n = 0; n < 8; ++n) acc[n] = zero;

    const float* Aagg = msg + (long long)rowA * IN_DIM + 2 * hi;
    const float* Aroot = x  + (long long)rowA * IN_DIM + 2 * hi;
    const float* WlB = Wl + 2 * hi;
    const float* WrB = Wr + 2 * hi;

    // ---- branch 1: (msg * invd) @ Wl^T --------------------------------
    for (int k = 0; k < IN_DIM; k += 4) {
        v2f a = *(const v2f*)(Aagg + k);
        a *= invd;                                  // fold mean into A fragment
#pragma unroll
        for (int n = 0; n < 8; ++n) {
            v2f b = *(const v2f*)(WlB + (n * 16 + l15) * IN_DIM + k);
            acc[n] = __builtin_amdgcn_wmma_f32_16x16x4_f32(
                false, a, false, b, (short)0, acc[n], false, false);
        }
    }
    // ---- branch 2: x @ Wr^T (same accumulators) -----------------------
    for (int k = 0; k < IN_DIM; k += 4) {
        v2f a = *(const v2f*)(Aroot + k);
#pragma unroll
        for (int n = 0; n < 8; ++n) {
            v2f b = *(const v2f*)(WrB + (n * 16 + l15) * IN_DIM + k);
            acc[n] = __builtin_amdgcn_wmma_f32_16x16x4_f32(
                false, a, false, b, (short)0, acc[n], false, false);
        }
    }

    // ---- bias ----------------------------------------------------------
#pragma unroll
    for (int n = 0; n < 8; ++n) {
        float b = bl[n * 16 + l15];
#pragma unroll
        for (int j = 0; j < 8; ++j) acc[n][j] += b;
    }

    // ---- row-wise L2 normalize + store ---------------------------------
    // Row (j, hi) lives across the 16 lanes of one half-wave over 8 N-tiles;
    // xor-butterfly over masks 1,2,4,8 reduces within each 16-lane half.
#pragma unroll
    for (int j = 0; j < 8; ++j) {
        float s = 0.0f;
#pragma unroll
        for (int n = 0; n < 8; ++n) { float v = acc[n][j]; s += v * v; }
        s += __shfl_xor(s, 1, 32);
        s += __shfl_xor(s, 2, 32);
        s += __shfl_xor(s, 4, 32);
        s += __shfl_xor(s, 8, 32);
        float scale = 1.0f / fmaxf(sqrtf(s), 1e-12f);
        int m = m0 + j + 8 * hi;
        if (m < N) {
            float* orow = out + (long long)m * OUT_DIM + l15;
#pragma unroll
            for (int n = 0; n < 8; ++n) orow[n * 16] = acc[n][j] * scale;
        }
    }
}

// ---------------------------------------------------------------------------
extern "C" void kernel_launch(void* const* d_in, const int* in_sizes, int n_in,
                              void* d_out, int out_size, void* d_ws, size_t ws_size,
                              hipStream_t stream) {
    const float* x  = (const float*)d_in[0];
    const int*   ei = (const int*)  d_in[1];   // edge_index [2,E] (harness: integer -> int32)
    const float* Wl = (const float*)d_in[2];
    const float* bl = (const float*)d_in[3];
    const float* Wr = (const float*)d_in[4];
    float*       out = (float*)d_out;

    const int N = in_sizes[0] / IN_DIM;  // 100000
    const int E = in_sizes[1] / 2;       // 1600000

    float* msg = (float*)d_ws;                       // [N,128] = 51.2 MB
    float* deg = msg + (long long)N * IN_DIM;        // [N]     =  0.4 MB

    long long zn = (long long)N * IN_DIM + N;
    sage_zero_kernel<<<2048, 256, 0, stream>>>(msg, zn);
    sage_scatter_kernel<<<4096, 256, 0, stream>>>(x, ei, msg, deg, E);

    const int ntiles = (N + 15) / 16;                // 6250
    const int blocks = (ntiles + 7) / 8;             // 8 waves / block
    sage_gemm_kernel<<<blocks, 256, 0, stream>>>(x, msg, deg, Wl, bl, Wr, out, N, ntiles);
}